// FFEdgeCountingAutoencoder_19593640804420
// MI455X (gfx1250) — compile-verified
//
#include <hip/hip_runtime.h>
#include <stdint.h>

#define IN_F  512
#define HID   256
#define BATCH 1024
#define MASK0_WORDS (HID * (IN_F / 32))   // 256*16 = 4096 words (16 KB)
#define MASK1_WORDS (IN_F * (HID / 32))   // 512*8  = 4096 words (16 KB)

// g(u) = -log(-log(clip(u, 1e-10, 1-1e-10))), computed in f32 like the reference.
__device__ __forceinline__ float gumbel_from_uniform(float u) {
    u = fminf(fmaxf(u, 1e-10f), 1.0f - 1e-10f);
    return -logf(-logf(u));
}

// Pack the hard Gumbel-softmax edge decisions into bitmasks.
// bit == 1  <=>  argmax(logits + g) == 1  <=>  (l1+g1) > (l0+g0)  (argmax ties -> index 0)
__global__ __launch_bounds__(256)
void gen_masks_kernel(const float* __restrict__ logits0, const float* __restrict__ u0,
                      const float* __restrict__ logits1, const float* __restrict__ u1,
                      uint32_t* __restrict__ mask0, uint32_t* __restrict__ mask1) {
    const uint32_t idx = blockIdx.x * 256u + threadIdx.x;
    const uint32_t N0  = HID * IN_F;  // 131072 (o,i) pairs for layer 0
    float z0, z1;
    uint32_t* dst;
    uint32_t  widx;
    if (idx < N0) {
        const float2 l = ((const float2*)logits0)[idx];
        const float2 u = ((const float2*)u0)[idx];
        z0 = l.x + gumbel_from_uniform(u.x);
        z1 = l.y + gumbel_from_uniform(u.y);
        dst = mask0; widx = idx >> 5;
    } else {
        const uint32_t j = idx - N0;
        const float2 l = ((const float2*)logits1)[j];
        const float2 u = ((const float2*)u1)[j];
        z0 = l.x + gumbel_from_uniform(u.x);
        z1 = l.y + gumbel_from_uniform(u.y);
        dst = mask1; widx = j >> 5;
    }
    // wave32 ballot: bit k = lane k's decision; idx is 32-aligned per wave.
    const uint32_t word = __builtin_amdgcn_ballot_w32(z1 > z0);
    if ((threadIdx.x & 31u) == 0u) dst[widx] = word;
}

// One block per batch row. 256 threads.
// Layer 0: h[o] = max(0, min_i (mask0[o,i] ? x[i] : 1.0))   (o = tid)
// Layer 1: out[o] = max_i (mask1[o,i] ? h[i] : 0)           (o = tid, tid+256)
__global__ __launch_bounds__(256)
void forward_kernel(const float* __restrict__ x,
                    const uint32_t* __restrict__ mask0,
                    const uint32_t* __restrict__ mask1,
                    float* __restrict__ out) {
    __shared__ __align__(16) float xs[IN_F];  // 2 KB
    __shared__ __align__(16) float hs[HID];   // 1 KB
    const uint32_t tid = threadIdx.x;
    const uint32_t b   = blockIdx.x;

    // ---- Stage x row into LDS with the CDNA5 async DMA path (ASYNCcnt). ----
    // GVS mode: saddr = row base (SGPR pair), vaddr = 32-bit per-lane byte offset,
    // vdst = LDS byte address. 8 bytes per lane -> 512 floats per block.
    {
        const float* xrow   = x + (size_t)b * IN_F;
        uint32_t lds_dst    = (uint32_t)(uintptr_t)(&xs[0]) + tid * 8u;
        uint32_t voff       = tid * 8u;
        asm volatile(
            "global_load_async_to_lds_b64 %0, %1, %2\n\t"
            "s_wait_asynccnt 0"
            :
            : "v"(lds_dst), "v"(voff), "s"(xrow)
            : "memory");
    }
    // Pull this thread's mask0 row toward the caches while we sync.
    __builtin_prefetch(mask0 + tid * (IN_F / 32), 0, 3);
    __syncthreads();

    // ---------------- Layer 0: masked min over 512 inputs ----------------
    {
        const uint4*  mrow = (const uint4*)(mask0 + tid * (IN_F / 32)); // 16 words
        const float4* xs4  = (const float4*)xs;
        float acc = __builtin_inff();
#pragma unroll
        for (int q = 0; q < 4; ++q) {
            const uint4 mw = mrow[q];
            const uint32_t w[4] = {mw.x, mw.y, mw.z, mw.w};
#pragma unroll
            for (int r = 0; r < 4; ++r) {
                const uint32_t m = w[r];
#pragma unroll
                for (int v = 0; v < 8; ++v) {
                    const float4 xv = xs4[(q * 4 + r) * 8 + v];  // LDS b128 broadcast
                    const int s = v * 4;
                    acc = fminf(acc, ((m >> (s + 0)) & 1u) ? xv.x : 1.0f);
                    acc = fminf(acc, ((m >> (s + 1)) & 1u) ? xv.y : 1.0f);
                    acc = fminf(acc, ((m >> (s + 2)) & 1u) ? xv.z : 1.0f);
                    acc = fminf(acc, ((m >> (s + 3)) & 1u) ? xv.w : 1.0f);
                }
            }
        }
        hs[tid] = fmaxf(0.0f, acc);  // max(0, .) commutes with min (monotone)
    }
    __syncthreads();

    // ---------------- Layer 1: masked max over 256 hidden ----------------
    const float4* hs4 = (const float4*)hs;
    float res[2];
#pragma unroll
    for (int half = 0; half < 2; ++half) {
        const uint32_t o   = tid + half * 256u;
        const uint4* mrow  = (const uint4*)(mask1 + o * (HID / 32)); // 8 words
        float acc = 0.0f;  // unselected edges contribute exactly 0
#pragma unroll
        for (int q = 0; q < 2; ++q) {
            const uint4 mw = mrow[q];
            const uint32_t w[4] = {mw.x, mw.y, mw.z, mw.w};
#pragma unroll
            for (int r = 0; r < 4; ++r) {
                const uint32_t m = w[r];
#pragma unroll
                for (int v = 0; v < 8; ++v) {
                    const float4 hv = hs4[(q * 4 + r) * 8 + v];  // LDS b128 broadcast
                    const int s = v * 4;
                    acc = fmaxf(acc, ((m >> (s + 0)) & 1u) ? hv.x : 0.0f);
                    acc = fmaxf(acc, ((m >> (s + 1)) & 1u) ? hv.y : 0.0f);
                    acc = fmaxf(acc, ((m >> (s + 2)) & 1u) ? hv.z : 0.0f);
                    acc = fmaxf(acc, ((m >> (s + 3)) & 1u) ? hv.w : 0.0f);
                }
            }
        }
        res[half] = acc;
    }
    out[(size_t)b * IN_F + tid]        = res[0];
    out[(size_t)b * IN_F + tid + 256u] = res[1];
}

extern "C" void kernel_launch(void* const* d_in, const int* in_sizes, int n_in,
                              void* d_out, int out_size, void* d_ws, size_t ws_size,
                              hipStream_t stream) {
    const float* x       = (const float*)d_in[0];
    const float* logits0 = (const float*)d_in[1];
    const float* u0      = (const float*)d_in[2];
    const float* logits1 = (const float*)d_in[3];
    const float* u1      = (const float*)d_in[4];

    uint32_t* mask0 = (uint32_t*)d_ws;              // 16 KB
    uint32_t* mask1 = mask0 + MASK0_WORDS;          // 16 KB  (total ws use: 32 KB)

    // 262144 edge decisions -> 1024 blocks of 256 threads
    gen_masks_kernel<<<(HID * IN_F + IN_F * HID) / 256, 256, 0, stream>>>(
        logits0, u0, logits1, u1, mask0, mask1);

    // One block per batch row
    forward_kernel<<<BATCH, 256, 0, stream>>>(x, mask0, mask1, (float*)d_out);
}